// CTC_72146860638614
// MI455X (gfx1250) — compile-verified
//
#include <hip/hip_runtime.h>

// Problem constants (fixed by reference)
#define TT 2000
#define BB 64
#define VV 128
#define LL 100
#define SS (2 * LL + 1)   // 201 extended states
#define NEGV (-1e30f)
#define SPL 7             // states per lane: 32*7 = 224 >= 201

typedef __attribute__((ext_vector_type(2))) float v2f;
typedef __attribute__((ext_vector_type(8))) float v8f;

// ---------------------------------------------------------------------------
// Kernel 1: per-(t,b) logsumexp over V=128, WMMA-accelerated sum of exp.
// One wave32 handles 16 rows. A-matrix 16x4 f32 layout: VGPR0 = K0 (lanes 0-15)
// / K2 (lanes 16-31), VGPR1 = K1 / K3. B = all-ones 4x16 -> D[m,n] = row sums
// replicated; 32 accumulating WMMAs cover K=128. All adds exact f32.
// ---------------------------------------------------------------------------
__global__ __launch_bounds__(128) void ctc_lse_kernel(
    const float* __restrict__ x, float* __restrict__ lse) {
  const int wid  = threadIdx.x >> 5;
  const int lane = threadIdx.x & 31;
  const int group = blockIdx.x * 4 + wid;     // 8000 groups of 16 rows
  const int rbase = group << 4;
  const int mrow = lane & 15;
  const int half = lane >> 4;
  const float* rp = x + (size_t)(rbase + mrow) * VV + half * 2;

  // Pass 1: row max (lanes r and r+16 cover disjoint K of the same row)
  float mx = -3.4e38f;
#pragma unroll 8
  for (int k = 0; k < VV; k += 4) {
    mx = fmaxf(mx, fmaxf(rp[k], rp[k + 1]));
  }
  mx = fmaxf(mx, __shfl_xor(mx, 16, 32));     // full row max on both halves

  // Pass 2: sum of exp via WMMA (L2 hits; exact f32 accumulation)
  v8f acc = {0.f, 0.f, 0.f, 0.f, 0.f, 0.f, 0.f, 0.f};
  v2f bones;
  bones.x = 1.0f;
  bones.y = 1.0f;
#pragma unroll 4
  for (int k = 0; k < VV; k += 4) {
    v2f A;
    A.x = __expf(rp[k] - mx);
    A.y = __expf(rp[k + 1] - mx);
    acc = __builtin_amdgcn_wmma_f32_16x16x4_f32(
        /*neg_a=*/false, A, /*neg_b=*/false, bones,
        /*c_mod=*/(short)0, acc, /*reuse_a=*/false, /*reuse_b=*/false);
  }

  // D layout: VGPR r, lanes 0-15 = sum(row r); lanes 16-31 = sum(row 8+r).
  float oth[8];
#pragma unroll
  for (int r = 0; r < 8; ++r) oth[r] = __shfl_xor(acc[r], 16, 32);

  float sel = acc[0];
#pragma unroll
  for (int r = 1; r < 8; ++r) sel = (mrow == r) ? acc[r] : sel;
#pragma unroll
  for (int r = 0; r < 8; ++r) sel = (mrow == 8 + r) ? oth[r] : sel;

  if (lane < 16) lse[rbase + lane] = mx + __logf(sel);
}

// ---------------------------------------------------------------------------
// Kernel 2: CTC alpha recursion. One wave32 per batch element; 7 consecutive
// states per lane live in registers. Per timestep: 2 __shfl_up for the
// shift-by-1/2 dependencies, 7 logaddexp3 chains, 7 gather loads from the
// current frame's 512B row (L2-resident), prefetch of the next row.
// No barriers in the hot loop.
// ---------------------------------------------------------------------------
__global__ __launch_bounds__(32) void ctc_alpha_kernel(
    const float* __restrict__ x, const int* __restrict__ target,
    const int* __restrict__ sizes, const int* __restrict__ label_lens,
    const float* __restrict__ lse, float* __restrict__ ll) {
  const int b = blockIdx.x;
  const int lane = threadIdx.x;

  // Per-lane extended-label metadata: ext index + skip-transition legality
  int idx[SPL];
  bool skipok[SPL];
#pragma unroll
  for (int j = 0; j < SPL; ++j) {
    const int s = lane * SPL + j;
    int id = 0;
    bool ok = false;
    if (s < SS && (s & 1)) {
      id = target[b * LL + ((s - 1) >> 1)];
      const int prev = (s >= 3) ? target[b * LL + ((s - 3) >> 1)] : -1;
      ok = (id != 0) && (id != prev);
    }
    idx[j] = id;
    skipok[j] = ok;
  }

  // t = 0 init: states 0 (blank) and 1 (first label)
  float a[SPL];
  {
    const float* r0 = x + (size_t)b * VV;
    const float l0 = lse[b];
#pragma unroll
    for (int j = 0; j < SPL; ++j) {
      const int s = lane * SPL + j;
      float v = NEGV;
      if (s == 0) v = r0[0] - l0;
      else if (s == 1) v = r0[idx[j]] - l0;
      a[j] = v;
    }
  }

  // Alpha is frozen for t >= sizes[b] in the reference -> just stop there.
  int tEnd = sizes[b];
  if (tEnd > TT) tEnd = TT;

  for (int t = 1; t < tEnd; ++t) {
    const float* rp = x + ((size_t)t * BB + b) * VV;
    const float lt = lse[t * BB + b];

    // Prefetch next timestep's 512B row (32 lanes x 16B spacing covers it)
    if (t + 1 < TT)
      __builtin_prefetch(rp + (size_t)BB * VV + lane * 4, 0, 1);

    float up6 = __shfl_up(a[6], 1, 32);   // alpha[7*lane - 1]
    float up5 = __shfl_up(a[5], 1, 32);   // alpha[7*lane - 2]
    if (lane == 0) { up6 = NEGV; up5 = NEGV; }

    float na[SPL];
#pragma unroll
    for (int j = 0; j < SPL; ++j) {
      const float a0 = a[j];
      const float a1 = (j == 0) ? up6 : a[j - 1];
      float a2 = (j == 0) ? up5 : ((j == 1) ? up6 : a[j - 2]);
      a2 = skipok[j] ? a2 : NEGV;
      const float m = fmaxf(a0, fmaxf(a1, a2));
      const float ssum = __expf(a0 - m) + __expf(a1 - m) + __expf(a2 - m);
      na[j] = m + __logf(ssum) + rp[idx[j]] - lt;
    }
#pragma unroll
    for (int j = 0; j < SPL; ++j) a[j] = na[j];
  }

  // Log-likelihood ends at states 2*len and 2*len-1
  __shared__ float fin[2];
  const int len = label_lens[b];
  const int s1 = 2 * len;
  const int s2 = 2 * len - 1;
#pragma unroll
  for (int j = 0; j < SPL; ++j) {
    const int s = lane * SPL + j;
    if (s == s1) fin[0] = a[j];
    if (s == s2) fin[1] = a[j];
  }
  __syncthreads();
  if (lane == 0) {
    const float xv = fin[0], yv = fin[1];
    const float m = fmaxf(xv, yv);
    ll[b] = m + __logf(__expf(xv - m) + __expf(yv - m));
  }
}

// ---------------------------------------------------------------------------
// Kernel 3: deterministic fixed-order batch sum, out = -sum(ll)
// ---------------------------------------------------------------------------
__global__ void ctc_sum_kernel(const float* __restrict__ ll,
                               float* __restrict__ out) {
  if (threadIdx.x == 0 && blockIdx.x == 0) {
    float s = 0.f;
    for (int b = 0; b < BB; ++b) s += ll[b];
    out[0] = -s;
  }
}

extern "C" void kernel_launch(void* const* d_in, const int* in_sizes, int n_in,
                              void* d_out, int out_size, void* d_ws,
                              size_t ws_size, hipStream_t stream) {
  (void)in_sizes; (void)n_in; (void)out_size; (void)ws_size;
  const float* x          = (const float*)d_in[0];  // (T,B,V) f32 logits
  const int*   target     = (const int*)d_in[1];    // (B*L) i32 labels
  const int*   sizes      = (const int*)d_in[2];    // (B) i32
  const int*   label_lens = (const int*)d_in[3];    // (B) i32
  float* out = (float*)d_out;

  float* ws  = (float*)d_ws;
  float* lse = ws;            // T*B floats = 512 KB
  float* ll  = ws + TT * BB;  // B floats

  // 8000 waves x 16 rows = 128000 logsumexp rows
  ctc_lse_kernel<<<(TT * BB) / 64, 128, 0, stream>>>(x, lse);
  // one wave32 per batch element
  ctc_alpha_kernel<<<BB, 32, 0, stream>>>(x, target, sizes, label_lens, lse, ll);
  ctc_sum_kernel<<<1, 32, 0, stream>>>(ll, out);
}